// SymmetricChannel_9680856285944
// MI455X (gfx1250) — compile-verified
//
#include <hip/hip_runtime.h>

// SymmetricChannel forward: out[b,l,:] = corrupt ? one_hot(rsym) : message[b,l,:]
// Memory-bound streaming kernel: ~258 MiB traffic, ~11 us floor at 23.3 TB/s.
// Layout: 1 wave32 per vocab row (128 f32 = 32 lanes x float4), 4 rows per wave.

typedef float v4f __attribute__((ext_vector_type(4)));

#define VOCAB          128
#define PROB           0.1f
#define WAVES_PER_BLK  8
#define ROWS_PER_WAVE  4

__global__ __launch_bounds__(256, 1) void symmetric_channel_kernel(
    const float* __restrict__ msg,
    const float* __restrict__ rand_u,
    const int*   __restrict__ rep,
    float*       __restrict__ out,
    int n_rows)
{
    const int lane = threadIdx.x & 31;
    const int wave = threadIdx.x >> 5;
    const int row0 = (blockIdx.x * WAVES_PER_BLK + wave) * ROWS_PER_WAVE;
    const int base = lane * 4;   // first vocab index held by this lane

    // ---- Issue all memory up front: 4x global_load_b128 (NT) + scalar side data.
    v4f   v[ROWS_PER_WAVE];
    float ru[ROWS_PER_WAVE];
    int   rp[ROWS_PER_WAVE];
#pragma unroll
    for (int k = 0; k < ROWS_PER_WAVE; ++k) {
        const int r = row0 + k;
        if (r < n_rows) {
            const v4f* src = (const v4f*)(msg + (size_t)r * VOCAB);
            v[k]  = __builtin_nontemporal_load(src + lane);  // touch-once: TH=NT
            ru[k] = rand_u[r];   // same address across wave -> 1 coalesced request
            rp[k] = rep[r];
        } else {
            v[k] = (v4f)(0.0f); ru[k] = 1.0f; rp[k] = 0;
        }
    }

    // ---- Process rows in load-issue order (compiler uses partial loadcnt waits).
#pragma unroll
    for (int k = 0; k < ROWS_PER_WAVE; ++k) {
        const int r = row0 + k;
        if (r >= n_rows) continue;
        const v4f val = v[k];

        // Lane-local argmax over 4 elements; strict '>' keeps lowest index on tie
        // (matches jnp.argmax first-occurrence semantics).
        float bv = val.x; int bi = base;
        if (val.y > bv) { bv = val.y; bi = base + 1; }
        if (val.z > bv) { bv = val.z; bi = base + 2; }
        if (val.w > bv) { bv = val.w; bi = base + 3; }

        // wave32 butterfly reduction: (max value, min index among maxima).
#pragma unroll
        for (int off = 16; off > 0; off >>= 1) {
            float ov = __shfl_xor(bv, off, 32);
            int   oi = __shfl_xor(bi, off, 32);
            if (ov > bv || (ov == bv && oi < bi)) { bv = ov; bi = oi; }
        }
        const int m = bi;                       // argmax symbol (wave-uniform)

        // Excluded-category replacement symbol via offset shift.
        const bool corrupt = (ru[k] < PROB) && (m != 0);  // wave-uniform branch
        const int  msg_exp = (m != 0) ? m : 1;
        const int  r1      = rp[k] + 1;
        const int  rsym    = r1 + ((r1 >= msg_exp) ? 1 : 0);

        v4f o;
        if (corrupt) {
            o.x = (base + 0 == rsym) ? 1.0f : 0.0f;
            o.y = (base + 1 == rsym) ? 1.0f : 0.0f;
            o.z = (base + 2 == rsym) ? 1.0f : 0.0f;
            o.w = (base + 3 == rsym) ? 1.0f : 0.0f;
        } else {
            o = val;
        }

        v4f* dst = (v4f*)(out + (size_t)r * VOCAB);
        __builtin_nontemporal_store(o, dst + lane);  // streaming store, TH=NT
    }
}

extern "C" void kernel_launch(void* const* d_in, const int* in_sizes, int n_in,
                              void* d_out, int out_size, void* d_ws, size_t ws_size,
                              hipStream_t stream) {
    const float* msg    = (const float*)d_in[0];   // [B, L, V] f32
    const float* rand_u = (const float*)d_in[1];   // [B, L]    f32
    const int*   rep    = (const int*)  d_in[2];   // [B, L]    i32
    float*       out    = (float*)d_out;           // [B, L, V] f32

    const int n_rows = in_sizes[1];                // B*L = 262144
    const int rows_per_block = WAVES_PER_BLK * ROWS_PER_WAVE;  // 32
    const int grid = (n_rows + rows_per_block - 1) / rows_per_block;

    symmetric_channel_kernel<<<grid, 256, 0, stream>>>(msg, rand_u, rep, out, n_rows);
}